// GAT_ancestor_84817014161574
// MI455X (gfx1250) — compile-verified
//
#include <hip/hip_runtime.h>
#include <math.h>

typedef __attribute__((ext_vector_type(2))) float v2f;
typedef __attribute__((ext_vector_type(8))) float v8f;

#define NNODES 100000
#define NEDGES 1600000
#define NEG_SLOPE 0.2f

// ---------- ordered-float encoding for atomicMax on floats ----------
__device__ __forceinline__ unsigned fkey(float f) {
  unsigned u = __float_as_uint(f);
  return (u & 0x80000000u) ? ~u : (u | 0x80000000u);
}
__device__ __forceinline__ float fkey_inv(unsigned k) {
  unsigned u = (k & 0x80000000u) ? (k & 0x7FFFFFFFu) : ~k;
  return __uint_as_float(u);
}

// ---------- fp32 WMMA GEMM: H[N,F] = A[N,K] @ W[K,F] ----------
// one wave per 16x16 output tile; K stepped by 4 via V_WMMA_F32_16X16X4_F32
__global__ void wmma_gemm_f32(const float* __restrict__ A, int K,
                              const float* __restrict__ W, int F,
                              float* __restrict__ H,
                              int rowTiles, int colTiles) {
  int wave = (blockIdx.x * blockDim.x + threadIdx.x) >> 5;
  int lane = threadIdx.x & 31;
  if (wave >= rowTiles * colTiles) return;   // wave-uniform guard (EXEC all-1 below)
  int rt = wave / colTiles;
  int ct = wave - rt * colTiles;
  int kh  = lane >> 4;   // half-wave selects K pair {0,1} vs {2,3}
  int l16 = lane & 15;
  int row = rt * 16 + l16;   // A: M = l16 in both halves
  int col = ct * 16 + l16;   // B/D: N = l16 in both halves
  v8f acc = {};
  for (int k0 = 0; k0 < K; k0 += 4) {
    int ka = k0 + 2 * kh;
    v2f a, b;
    a.x = A[row * K + ka];
    a.y = A[row * K + ka + 1];
    b.x = W[ka * F + col];
    b.y = W[(ka + 1) * F + col];
    acc = __builtin_amdgcn_wmma_f32_16x16x4_f32(
        /*neg_a=*/false, a, /*neg_b=*/false, b,
        /*c_mod=*/(short)0, acc, /*reuse_a=*/false, /*reuse_b=*/false);
  }
#pragma unroll
  for (int r = 0; r < 8; ++r) {
    int m = r + 8 * kh;                       // C/D layout: lanes16-31 hold M=8..15
    H[(rt * 16 + m) * F + col] = acc[r];
  }
}

// ---------- per-node attention logits alpha_s = h@a_src, alpha_d = h@a_dst ----------
__global__ void alpha_kernel(const float* __restrict__ h,
                             const float* __restrict__ a_src,
                             const float* __restrict__ a_dst,
                             float* __restrict__ as_, float* __restrict__ ad_,
                             int n, int F) {
  int i = blockIdx.x * blockDim.x + threadIdx.x;
  if (i >= n) return;
  float s = 0.f, d = 0.f;
  for (int f = 0; f < F; ++f) {
    float v = h[i * F + f];
    s += v * a_src[f];
    d += v * a_dst[f];
  }
  as_[i] = s;
  ad_[i] = d;
}

// ---------- reset agg/denom/max buffers ----------
__global__ void reset_kernel(float* __restrict__ agg, float* __restrict__ denom,
                             unsigned* __restrict__ mx, int n, int F) {
  int idx = blockIdx.x * blockDim.x + threadIdx.x;
  if (idx < n * F) agg[idx] = 0.f;
  if (idx < n) {
    denom[idx] = 0.f;
    mx[idx] = 0x007FFFFFu;   // fkey(-inf)
  }
}

// ---------- pass 1: segment max of leaky_relu(alpha_s[src]+alpha_d[dst]) over dst ----------
__global__ void edge_max_kernel(const int* __restrict__ src, const int* __restrict__ dst,
                                int ne, int n,
                                const float* __restrict__ as_, const float* __restrict__ ad_,
                                unsigned* __restrict__ mx) {
  int e = blockIdx.x * blockDim.x + threadIdx.x;
  if (e >= ne + n) return;
  int s = (e < ne) ? src[e] : (e - ne);   // self-loops appended
  int d = (e < ne) ? dst[e] : (e - ne);
  float t = as_[s] + ad_[d];
  t = t > 0.f ? t : NEG_SLOPE * t;
  atomicMax(&mx[d], fkey(t));
}

// ---------- pass 2: one wave per edge; accumulate exp(e-m) and exp(e-m)*h[src] ----------
__global__ void edge_acc_kernel(const int* __restrict__ src, const int* __restrict__ dst,
                                int ne, int n,
                                const float* __restrict__ as_, const float* __restrict__ ad_,
                                const unsigned* __restrict__ mx,
                                const float* __restrict__ h, int F,
                                float* __restrict__ agg, float* __restrict__ denom) {
  int wid = (blockIdx.x * blockDim.x + threadIdx.x) >> 5;
  int lane = threadIdx.x & 31;
  if (wid >= ne + n) return;
  int s = (wid < ne) ? src[wid] : (wid - ne);
  int d = (wid < ne) ? dst[wid] : (wid - ne);
  float t = as_[s] + ad_[d];
  t = t > 0.f ? t : NEG_SLOPE * t;
  float w = expf(t - fkey_inv(mx[d]));   // <= 1, no overflow
  if (lane == 0) atomicAdd(&denom[d], w);
  for (int f = lane; f < F; f += 32)
    atomicAdd(&agg[d * F + f], w * h[s * F + f]);
}

// ---------- normalize, add bias, optional ELU, write to one or two destinations ----------
__global__ void finalize_kernel(const float* __restrict__ agg, const float* __restrict__ denom,
                                const float* __restrict__ bias, int n, int F,
                                float* __restrict__ out1, int ld1, int off1,
                                float* __restrict__ out2, int ld2, int off2, int do_elu) {
  int idx = blockIdx.x * blockDim.x + threadIdx.x;
  if (idx >= n * F) return;
  int i = idx / F, f = idx - i * F;
  float v = agg[idx] / (denom[i] + 1e-16f) + bias[f];
  if (do_elu) v = v > 0.f ? v : (expf(v) - 1.f);
  out1[i * ld1 + off1 + f] = v;
  if (out2) out2[i * ld2 + off2 + f] = v;
}

// ---------- copy x2 into concat buffer columns ----------
__global__ void copy_cols_kernel(const float* __restrict__ x, int Fx,
                                 float* __restrict__ dst, int ldd, int off, int n) {
  int idx = blockIdx.x * blockDim.x + threadIdx.x;
  if (idx >= n * Fx) return;
  int i = idx / Fx, f = idx - i * Fx;
  dst[i * ldd + off + f] = x[idx];
}

// ---------- BN statistics: one block per channel, double LDS reduction ----------
__global__ void bn_stats_kernel(const float* __restrict__ logits, int n,
                                float* __restrict__ mu, float* __restrict__ rsig) {
  int c = blockIdx.x;       // 16 blocks
  int t = threadIdx.x;      // 256 threads
  double s = 0.0, q = 0.0;
  for (int i = t; i < n; i += 256) {
    float v = logits[i * 16 + c];
    s += v;
    q += (double)v * (double)v;
  }
  __shared__ double ss[256], qq[256];
  ss[t] = s; qq[t] = q;
  __syncthreads();
  for (int o = 128; o > 0; o >>= 1) {
    if (t < o) { ss[t] += ss[t + o]; qq[t] += qq[t + o]; }
    __syncthreads();
  }
  if (t == 0) {
    double mean = ss[0] / n;
    double var = qq[0] / n - mean * mean;   // biased, matches jnp.var
    mu[c] = (float)mean;
    rsig[c] = rsqrtf((float)var + 1e-5f);
  }
}

// ---------- fused BN affine + log_softmax ----------
__global__ void bn_logsoftmax_kernel(const float* __restrict__ logits,
                                     const float* __restrict__ mu, const float* __restrict__ rsig,
                                     const float* __restrict__ gamma, const float* __restrict__ beta,
                                     float* __restrict__ out, int n) {
  int i = blockIdx.x * blockDim.x + threadIdx.x;
  if (i >= n) return;
  float y[16];
  float mx = -1e30f;
#pragma unroll
  for (int c = 0; c < 16; ++c) {
    float v = (logits[i * 16 + c] - mu[c]) * rsig[c] * gamma[c] + beta[c];
    y[c] = v;
    mx = fmaxf(mx, v);
  }
  float se = 0.f;
#pragma unroll
  for (int c = 0; c < 16; ++c) se += expf(y[c] - mx);
  float lse = mx + logf(se);
#pragma unroll
  for (int c = 0; c < 16; ++c) out[i * 16 + c] = y[c] - lse;
}

extern "C" void kernel_launch(void* const* d_in, const int* in_sizes, int n_in,
                              void* d_out, int out_size, void* d_ws, size_t ws_size,
                              hipStream_t stream) {
  const int N = NNODES, E = NEDGES;
  // inputs per setup_inputs() order
  const float* x1    = (const float*)d_in[0];
  const float* x2    = (const float*)d_in[1];
  const int*   ei1   = (const int*)d_in[2];   // [0..E)=src, [E..2E)=dst
  const int*   ei2   = (const int*)d_in[3];
  const float* W0    = (const float*)d_in[4];
  const float* asrc0 = (const float*)d_in[5];
  const float* adst0 = (const float*)d_in[6];
  const float* b0    = (const float*)d_in[7];
  const float* W1    = (const float*)d_in[8];
  const float* asrc1 = (const float*)d_in[9];
  const float* adst1 = (const float*)d_in[10];
  const float* b1    = (const float*)d_in[11];
  const float* W2    = (const float*)d_in[12];
  const float* asrc2 = (const float*)d_in[13];
  const float* adst2 = (const float*)d_in[14];
  const float* b2    = (const float*)d_in[15];
  const float* gamma = (const float*)d_in[16];
  const float* beta  = (const float*)d_in[17];

  // workspace carving (everything fits in MI455X's 192MB L2 once touched)
  char* ws = (char*)d_ws;
  size_t off = 0;
  auto carve = [&](size_t bytes) -> char* {
    char* p = ws + off;
    off = (off + bytes + 255) & ~(size_t)255;
    return p;
  };
  float*    xin1   = (float*)carve((size_t)N * 128 * 4);   // [h0 | x2]
  float*    xend   = (float*)carve((size_t)N * 128 * 4);   // [h0 | h1]
  float*    h_tmp  = (float*)carve((size_t)N * 64 * 4);    // pre-aggregation h = A@W
  float*    agg    = (float*)carve((size_t)N * 64 * 4);    // unnormalized softmax-weighted sum
  float*    alpha_s= (float*)carve((size_t)N * 4);
  float*    alpha_d= (float*)carve((size_t)N * 4);
  unsigned* maxbuf = (unsigned*)carve((size_t)N * 4);
  float*    denom  = (float*)carve((size_t)N * 4);
  float*    logits = (float*)carve((size_t)N * 16 * 4);
  float*    mu     = (float*)carve(16 * 4);
  float*    rsig   = (float*)carve(16 * 4);
  (void)ws_size; (void)n_in; (void)in_sizes; (void)out_size;

  const int TB = 256;
  auto cdiv = [](long long a, long long b) -> int { return (int)((a + b - 1) / b); };

  auto run_layer = [&](const float* A, int K, const float* W, int F, const int* eidx,
                       const float* asrc, const float* adst, const float* bias,
                       float* out1, int ld1, int o1,
                       float* out2, int ld2, int o2, int do_elu) {
    int rowTiles = N / 16, colTiles = F / 16;
    int totalTiles = rowTiles * colTiles;
    wmma_gemm_f32<<<cdiv((long long)totalTiles * 32, TB), TB, 0, stream>>>(
        A, K, W, F, h_tmp, rowTiles, colTiles);
    alpha_kernel<<<cdiv(N, TB), TB, 0, stream>>>(h_tmp, asrc, adst, alpha_s, alpha_d, N, F);
    reset_kernel<<<cdiv((long long)N * F, TB), TB, 0, stream>>>(agg, denom, maxbuf, N, F);
    edge_max_kernel<<<cdiv((long long)E + N, TB), TB, 0, stream>>>(
        eidx, eidx + E, E, N, alpha_s, alpha_d, maxbuf);
    edge_acc_kernel<<<cdiv(((long long)E + N) * 32, TB), TB, 0, stream>>>(
        eidx, eidx + E, E, N, alpha_s, alpha_d, maxbuf, h_tmp, F, agg, denom);
    finalize_kernel<<<cdiv((long long)N * F, TB), TB, 0, stream>>>(
        agg, denom, bias, N, F, out1, ld1, o1, out2, ld2, o2, do_elu);
  };

  // fill xin1[:,64:128] = x2 (independent of layer 0; stream-ordered anyway)
  copy_cols_kernel<<<cdiv((long long)N * 64, TB), TB, 0, stream>>>(x2, 64, xin1, 128, 64, N);

  // layer 0: h0 = elu(gat(x1, ei1)) -> xin1[:, :64] and xend[:, :64]
  run_layer(x1, 64, W0, 64, ei1, asrc0, adst0, b0, xin1, 128, 0, xend, 128, 0, 1);
  // layer 1: h1 = elu(gat([h0|x2], ei2)) -> xend[:, 64:128]
  run_layer(xin1, 128, W1, 64, ei2, asrc1, adst1, b1, xend, 128, 64, nullptr, 0, 0, 1);
  // layer 2: logits = gat([h0|h1], ei2)   (no ELU)
  run_layer(xend, 128, W2, 16, ei2, asrc2, adst2, b2, logits, 16, 0, nullptr, 0, 0, 0);

  // BatchNorm (batch stats) + log_softmax
  bn_stats_kernel<<<16, 256, 0, stream>>>(logits, N, mu, rsig);
  bn_logsoftmax_kernel<<<cdiv(N, TB), TB, 0, stream>>>(
      logits, mu, rsig, gamma, beta, (float*)d_out, N);
}